// AudioOnlySpecAugment_40853728920174
// MI455X (gfx1250) — compile-verified
//
#include <hip/hip_runtime.h>
#include <stdint.h>

typedef float v4f __attribute__((ext_vector_type(4)));
typedef int   v4i __attribute__((vector_size(16)));   // matches builtin param type

#define GAS __attribute__((address_space(1)))
#define LAS __attribute__((address_space(3)))

namespace {
constexpr int kB  = 32;
constexpr int kT  = 2048;
constexpr int kD  = 1536;
constexpr int kA  = 1280;
constexpr int kV  = 256;     // kD - kA, pass-through channels
constexpr int kNT = 2;
constexpr int kNF = 2;
// int(1280 * 0.15) == 192 in double arithmetic (product rounds to exactly 192.0)
constexpr float kMaxFPlus1 = 193.0f;
constexpr int kThreads = 384; // 64 V-float4 lanes + 320 audio-float4 lanes
}

#if defined(__gfx1250__) && \
    __has_builtin(__builtin_amdgcn_global_load_async_to_lds_b128) && \
    __has_builtin(__builtin_amdgcn_global_store_async_from_lds_b128)
#define SPECAUG_ASYNC_V 1
#else
#define SPECAUG_ASYNC_V 0
#endif

__global__ __launch_bounds__(kThreads) void specaug_kernel(
    const float* __restrict__ X,
    const int*   __restrict__ lengths,
    const float* __restrict__ u_t,
    const float* __restrict__ u_t0,
    const float* __restrict__ u_f,
    const float* __restrict__ u_f0,
    float* __restrict__ out)
{
  __shared__ v4i vbuf[64];   // 1 KB staging for the pass-through V slab

  const int row = blockIdx.x;        // row = b * kT + t
  const int b   = row >> 11;         // / 2048
  const int t   = row & (kT - 1);
  const int tid = threadIdx.x;

  const size_t rowoff = (size_t)row * kD;
  const float* xrow = X + rowoff;
  float*       orow = out + rowoff;

  // ---- V slab: kick off async global->LDS DMA first so it overlaps the
  //      audio compute below (CDNA5 async engine, no VGPR data movement). ----
#if SPECAUG_ASYNC_V
  if (tid < 64) {
    GAS v4i* gsrc = (GAS v4i*)xrow + tid;
    LAS v4i* lptr = (LAS v4i*)&vbuf[tid];
    __builtin_amdgcn_global_load_async_to_lds_b128(gsrc, lptr, 0, 0);
  }
#endif

  // ---- Block-uniform mask scalars (lower to s_load / SALU). ----
  const int   len  = lengths[b];
  const float lenf = (float)len;
  const float maxt = floorf(lenf * 0.2f);   // TR = 0.2

  bool tmask = false;
#pragma unroll
  for (int n = 0; n < kNT; ++n) {
    const int tn  = (int)floorf(u_t[n * kB + b] * (maxt + 1.0f));
    const int rem = len - tn;
    const int t0  = (rem <= 0) ? 0
                  : (int)floorf(u_t0[n * kB + b] * ((float)rem + 1.0f));
    tmask |= (t >= t0) & (t < t0 + tn);
  }

  int f0a[kNF], f1a[kNF];
#pragma unroll
  for (int n = 0; n < kNF; ++n) {
    const int fn   = (int)floorf(u_f[n * kB + b] * kMaxFPlus1);
    const int f0m  = kA - fn;                 // fn <= 192, always positive
    const int f0   = (int)floorf(u_f0[n * kB + b] * ((float)f0m + 1.0f));
    f0a[n] = f0;
    f1a[n] = f0 + fn;
  }

  // ---- Audio slab: streaming b128 with NT hints; time-masked rows skip the
  //      read entirely (block-uniform branch). ----
  if (tid >= 64) {
    const int j  = tid - 64;     // audio float4 index, 0..319
    const int fb = j * 4;        // audio channel of component 0
    const GAS v4f* gx = (const GAS v4f*)(xrow + kV) + j;
    GAS v4f*       go = (GAS v4f*)(orow + kV) + j;
    if (tmask) {
      v4f z = (v4f)0.0f;
      __builtin_nontemporal_store(z, go);
    } else {
      v4f x = __builtin_nontemporal_load(gx);
      v4f r;
#pragma unroll
      for (int c = 0; c < 4; ++c) {
        const int f = fb + c;
        bool fm = false;
#pragma unroll
        for (int n = 0; n < kNF; ++n)
          fm |= (f >= f0a[n]) & (f < f1a[n]);
        r[c] = fm ? 0.0f : x[c];
      }
      __builtin_nontemporal_store(r, go);
    }
  }

  // ---- V slab: drain the async load, then DMA LDS -> global. ----
#if SPECAUG_ASYNC_V
#if __has_builtin(__builtin_amdgcn_s_wait_asynccnt)
  __builtin_amdgcn_s_wait_asynccnt(0);
#else
  asm volatile("s_wait_asynccnt 0" ::: "memory");
#endif
  if (tid < 64) {
    GAS v4i* gdst = (GAS v4i*)orow + tid;
    LAS v4i* lptr = (LAS v4i*)&vbuf[tid];
    __builtin_amdgcn_global_store_async_from_lds_b128(gdst, lptr, 0, 0);
  }
  // S_ENDPGM performs an implicit wait-idle, draining the async stores.
#else
  if (tid < 64) {
    const GAS v4f* gx = (const GAS v4f*)xrow + tid;
    GAS v4f*       go = (GAS v4f*)orow + tid;
    __builtin_nontemporal_store(__builtin_nontemporal_load(gx), go);
  }
#endif
}

extern "C" void kernel_launch(void* const* d_in, const int* in_sizes, int n_in,
                              void* d_out, int out_size, void* d_ws, size_t ws_size,
                              hipStream_t stream) {
  (void)in_sizes; (void)n_in; (void)out_size; (void)d_ws; (void)ws_size;
  const float* X       = (const float*)d_in[0];
  const int*   lengths = (const int*)  d_in[1];
  const float* u_t     = (const float*)d_in[2];
  const float* u_t0    = (const float*)d_in[3];
  const float* u_f     = (const float*)d_in[4];
  const float* u_f0    = (const float*)d_in[5];
  float* out = (float*)d_out;

  dim3 grid(kB * kT);   // one block per (b, t) row
  specaug_kernel<<<grid, kThreads, 0, stream>>>(X, lengths, u_t, u_t0, u_f, u_f0, out);
}